// SEM_39350490366351
// MI455X (gfx1250) — compile-verified
//
#include <hip/hip_runtime.h>
#include <math.h>

// Problem constants (match reference setup_inputs)
#define BB 8
#define NN 4096
#define DD 128
#define RR 512
#define TK 64

typedef float v2f __attribute__((ext_vector_type(2)));
typedef float v8f __attribute__((ext_vector_type(8)));

// ---------------------------------------------------------------------------
// WMMA f32 16x16x4 helpers (CDNA5 / gfx1250). One wave computes one 16x16 tile.
// A: MxK row-major, B: KxN row-major, C/D: 16x16 f32 (8 VGPRs).
// Lane layout per ISA 7.12.2:
//   A 16x4:  lanes 0-15 hold K=k0,k0+1; lanes 16-31 hold K=k0+2,k0+3 (row M=lane&15)
//   B 4x16:  v[0] = row k0 (lanes 0-15) / k0+2 (lanes 16-31); v[1] = k0+1 / k0+3
//   C/D:     VGPR r -> row m0+r (lanes 0-15), row m0+8+r (lanes 16-31)
// ---------------------------------------------------------------------------
__device__ __forceinline__ void wmma_kloop_f32(const float* __restrict__ A, int lda,
                                               const float* __restrict__ Bm, int ldb,
                                               int m0, int n0, int Kdim, v8f& acc) {
    const int lane = threadIdx.x & 31;
    const int half = lane >> 4;      // 0 or 1
    const int l    = lane & 15;
    const float* __restrict__ arow = A + (size_t)(m0 + l) * lda;
    const float* __restrict__ bcol = Bm + (n0 + l);
    for (int k0 = 0; k0 < Kdim; k0 += 4) {
        const int kk = k0 + half * 2;
        v2f a, b;
        a.x = arow[kk];
        a.y = arow[kk + 1];
        b.x = bcol[(size_t)kk * ldb];
        b.y = bcol[(size_t)(kk + 1) * ldb];
        acc = __builtin_amdgcn_wmma_f32_16x16x4_f32(false, a, false, b,
                                                    (short)0, acc, false, false);
    }
}

__device__ __forceinline__ v8f tile_load16(const float* __restrict__ C, int ldc,
                                           int m0, int n0) {
    const int lane = threadIdx.x & 31;
    const int half = lane >> 4;
    const int l    = lane & 15;
    v8f c;
#pragma unroll
    for (int r = 0; r < 8; ++r)
        c[r] = C[(size_t)(m0 + half * 8 + r) * ldc + n0 + l];
    return c;
}

__device__ __forceinline__ void tile_store16(float* __restrict__ Dst, int ldc,
                                             int m0, int n0, v8f c) {
    const int lane = threadIdx.x & 31;
    const int half = lane >> 4;
    const int l    = lane & 15;
#pragma unroll
    for (int r = 0; r < 8; ++r)
        Dst[(size_t)(m0 + half * 8 + r) * ldc + n0 + l] = c[r];
}

__device__ __forceinline__ float sigmoidf_(float x) { return 1.0f / (1.0f + expf(-x)); }

// ---------------------------------------------------------------------------
// 1) scorer = tanh(ht @ W_map + b_map); also inv-norm.  grid=B, block=128
// ---------------------------------------------------------------------------
__global__ __launch_bounds__(128)
void scorer_kernel(const float* __restrict__ ht, const float* __restrict__ W_map,
                   const float* __restrict__ b_map,
                   float* __restrict__ scorer_ws, float* __restrict__ scorer_out,
                   float* __restrict__ invn) {
    const int b = blockIdx.x, d = threadIdx.x;
    __shared__ float sh_ht[RR];
    for (int r = d; r < RR; r += 128) sh_ht[r] = ht[b * RR + r];
    __syncthreads();
    float acc = b_map[d];
    for (int r = 0; r < RR; ++r) acc += sh_ht[r] * W_map[r * DD + d];
    const float s = tanhf(acc);
    scorer_ws[b * DD + d]  = s;
    scorer_out[b * DD + d] = s;
    __shared__ float red[128];
    red[d] = s * s;
    __syncthreads();
    for (int off = 64; off > 0; off >>= 1) {
        if (d < off) red[d] += red[d + off];
        __syncthreads();
    }
    if (d == 0) invn[b] = rsqrtf(red[0]);
}

// ---------------------------------------------------------------------------
// 2) scores[b,n] = dot(node_embs[b,n,:], scorer[b,:]) * invn[b]
//    One wave32 per row; block=256 => 8 rows; grid = B*N/8 = 4096
// ---------------------------------------------------------------------------
__global__ __launch_bounds__(256)
void scores_kernel(const float* __restrict__ embs, const float* __restrict__ scorer,
                   const float* __restrict__ invn, float* __restrict__ scores) {
    const int wave = threadIdx.x >> 5, lane = threadIdx.x & 31;
    const int row = blockIdx.x * 8 + wave;            // 0..B*N-1
    const int b = row >> 12, n = row & (NN - 1);
    const float* __restrict__ x  = embs + ((size_t)b * NN + n) * DD;
    const float* __restrict__ sc = scorer + b * DD;
    float acc = 0.0f;
#pragma unroll
    for (int j = 0; j < 4; ++j) {
        const int d = lane + 32 * j;
        acc += x[d] * sc[d];
    }
    for (int m = 16; m > 0; m >>= 1) acc += __shfl_xor(acc, m, 32);
    if (lane == 0) scores[row] = acc * invn[b];
}

// ---------------------------------------------------------------------------
// 3) per-batch softmax stats (max, lse) + entropy.  grid=B, block=256
// ---------------------------------------------------------------------------
__global__ __launch_bounds__(256)
void stats_kernel(const float* __restrict__ scores, float* __restrict__ stats,
                  float* __restrict__ entropy_out) {
    const int b = blockIdx.x, t = threadIdx.x;
    const float* __restrict__ s = scores + (size_t)b * NN;
    __shared__ float red[256];
    __shared__ float shv[2];
    float m = -INFINITY;
    for (int i = t; i < NN; i += 256) m = fmaxf(m, s[i]);
    red[t] = m; __syncthreads();
    for (int off = 128; off > 0; off >>= 1) {
        if (t < off) red[t] = fmaxf(red[t], red[t + off]);
        __syncthreads();
    }
    if (t == 0) shv[0] = red[0];
    __syncthreads();
    const float M = shv[0];
    float S = 0.0f, T = 0.0f;
    for (int i = t; i < NN; i += 256) {
        const float e = expf(s[i] - M);
        S += e; T += e * s[i];
    }
    red[t] = S; __syncthreads();
    for (int off = 128; off > 0; off >>= 1) {
        if (t < off) red[t] += red[t + off];
        __syncthreads();
    }
    if (t == 0) shv[1] = red[0];
    __syncthreads();
    const float S0 = shv[1];
    red[t] = T; __syncthreads();
    for (int off = 128; off > 0; off >>= 1) {
        if (t < off) red[t] += red[t + off];
        __syncthreads();
    }
    if (t == 0) {
        const float T0 = red[0];
        const float lse = logf(S0);
        stats[b]      = M;
        stats[BB + b] = lse;
        entropy_out[b] = M + lse - T0 / S0;
    }
}

// ---------------------------------------------------------------------------
// 4) top-64 per batch via iterative argmax (descending, low-index tiebreak)
//    + policy_score = mean(topk vals) - M - lse.  grid=B, block=256
// ---------------------------------------------------------------------------
__global__ __launch_bounds__(256)
void topk_kernel(const float* __restrict__ scores, const float* __restrict__ stats,
                 int* __restrict__ tk, float* __restrict__ policy_out) {
    const int b = blockIdx.x, t = threadIdx.x;
    __shared__ float sv[NN];
    __shared__ float rv[256];
    __shared__ int   ri[256];
    __shared__ float sumv;
    for (int i = t; i < NN; i += 256) sv[i] = scores[(size_t)b * NN + i];
    if (t == 0) sumv = 0.0f;
    __syncthreads();
    for (int sel = 0; sel < TK; ++sel) {
        float bv = -INFINITY; int bi = NN;
        for (int i = t; i < NN; i += 256) {
            const float v = sv[i];
            if (v > bv || (v == bv && i < bi)) { bv = v; bi = i; }
        }
        rv[t] = bv; ri[t] = bi;
        __syncthreads();
        for (int off = 128; off > 0; off >>= 1) {
            if (t < off) {
                const float v2 = rv[t + off]; const int i2 = ri[t + off];
                if (v2 > rv[t] || (v2 == rv[t] && i2 < ri[t])) { rv[t] = v2; ri[t] = i2; }
            }
            __syncthreads();
        }
        if (t == 0) {
            tk[b * TK + sel] = ri[0];
            sumv += rv[0];
            sv[ri[0]] = -INFINITY;
        }
        __syncthreads();
    }
    if (t == 0) policy_out[b] = sumv / (float)TK - stats[b] - stats[BB + b];
}

// ---------------------------------------------------------------------------
// 5a) build z_topk (transposed z) [B,D,D] and ne [B,K,D].  grid=B, block=256
// ---------------------------------------------------------------------------
__global__ __launch_bounds__(256)
void gather_zne_kernel(const float* __restrict__ embs, const float* __restrict__ scores,
                       const int* __restrict__ tk,
                       float* __restrict__ z, float* __restrict__ ne) {
    const int b = blockIdx.x, t = threadIdx.x;
    __shared__ int   stk[TK];
    __shared__ float sts[TK];
    if (t < TK) {
        const int idx = tk[b * TK + t];
        stk[t] = idx;
        sts[t] = tanhf(scores[(size_t)b * NN + idx]);
    }
    __syncthreads();
    const float* __restrict__ eb = embs + (size_t)b * NN * DD;
    float* __restrict__ zb = z + (size_t)b * DD * DD;
    for (int e = t; e < DD * DD; e += 256) {
        const int d = e >> 7, i = e & 127;       // z_topk[d][i]
        const int k = i & 63;
        zb[e] = eb[(size_t)stk[k] * DD + d] * sts[k];
    }
    float* __restrict__ neb = ne + (size_t)b * TK * DD;
    for (int e = t; e < TK * DD; e += 256) {
        const int k = e >> 7, d = e & 127;
        neb[e] = eb[(size_t)stk[k] * DD + d];
    }
}

// ---------------------------------------------------------------------------
// 5b) gather A = Ahat[b, tk, tk], normalize by D^-1/2 both sides. grid=B, 256
// ---------------------------------------------------------------------------
__global__ __launch_bounds__(256)
void gather_A_kernel(const float* __restrict__ Ahat, const int* __restrict__ tk,
                     float* __restrict__ Anorm) {
    const int b = blockIdx.x, t = threadIdx.x;
    __shared__ float Ar[TK * TK];
    __shared__ float di[TK];
    __shared__ int   stk[TK];
    if (t < TK) stk[t] = tk[b * TK + t];
    __syncthreads();
    const float* __restrict__ Ab = Ahat + (size_t)b * NN * NN;
    for (int e = t; e < TK * TK; e += 256) {
        const int i = e >> 6, j = e & 63;
        Ar[e] = Ab[(size_t)stk[i] * NN + stk[j]];
    }
    __syncthreads();
    if (t < TK) {
        float s = 0.0f;
        for (int i = 0; i < TK; ++i) s += Ar[i * TK + t];
        di[t] = rsqrtf(s);
    }
    __syncthreads();
    float* __restrict__ Ao = Anorm + (size_t)b * TK * TK;
    for (int e = t; e < TK * TK; e += 256) {
        const int i = e >> 6, j = e & 63;
        Ao[e] = Ar[e] * di[i] * di[j];
    }
}

// ---------------------------------------------------------------------------
// 6) GRU gates: update = sigmoid(Wu Z + Uu Q + bu); RQ = sigmoid(Wr Z + Ur Q + br) * Q
//    grid = B*8 (tile rows), block=256 (8 waves = 8 tile cols)
// ---------------------------------------------------------------------------
__global__ __launch_bounds__(256)
void gru_gate_kernel(const float* __restrict__ Wu, const float* __restrict__ Uu,
                     const float* __restrict__ bu,
                     const float* __restrict__ Wr, const float* __restrict__ Ur,
                     const float* __restrict__ br,
                     const float* __restrict__ Z, const float* __restrict__ Q,
                     float* __restrict__ upd, float* __restrict__ rq) {
    const int b  = blockIdx.x >> 3;
    const int m0 = (blockIdx.x & 7) * 16;
    const int n0 = (threadIdx.x >> 5) * 16;
    const float* __restrict__ Zb = Z + (size_t)b * DD * DD;
    const float* __restrict__ Qb = Q + (size_t)b * DD * DD;

    v8f au = tile_load16(bu, DD, m0, n0);
    wmma_kloop_f32(Wu, DD, Zb, DD, m0, n0, DD, au);
    wmma_kloop_f32(Uu, DD, Qb, DD, m0, n0, DD, au);

    v8f ar = tile_load16(br, DD, m0, n0);
    wmma_kloop_f32(Wr, DD, Zb, DD, m0, n0, DD, ar);
    wmma_kloop_f32(Ur, DD, Qb, DD, m0, n0, DD, ar);

    v8f qt = tile_load16(Qb, DD, m0, n0);
    v8f u, rv;
#pragma unroll
    for (int r = 0; r < 8; ++r) {
        u[r]  = sigmoidf_(au[r]);
        rv[r] = sigmoidf_(ar[r]) * qt[r];
    }
    tile_store16(upd + (size_t)b * DD * DD, DD, m0, n0, u);
    tile_store16(rq  + (size_t)b * DD * DD, DD, m0, n0, rv);
}

// ---------------------------------------------------------------------------
// 7) h_cap = tanh(Wh Z + Uh RQ + bh); new_Q = (1-u)*Q + u*h_cap
// ---------------------------------------------------------------------------
__global__ __launch_bounds__(256)
void gru_out_kernel(const float* __restrict__ Wh, const float* __restrict__ Uh,
                    const float* __restrict__ bh,
                    const float* __restrict__ Z, const float* __restrict__ RQ,
                    const float* __restrict__ upd, const float* __restrict__ Q,
                    float* __restrict__ newQ) {
    const int b  = blockIdx.x >> 3;
    const int m0 = (blockIdx.x & 7) * 16;
    const int n0 = (threadIdx.x >> 5) * 16;
    const float* __restrict__ Zb  = Z  + (size_t)b * DD * DD;
    const float* __restrict__ RQb = RQ + (size_t)b * DD * DD;

    v8f ah = tile_load16(bh, DD, m0, n0);
    wmma_kloop_f32(Wh, DD, Zb,  DD, m0, n0, DD, ah);
    wmma_kloop_f32(Uh, DD, RQb, DD, m0, n0, DD, ah);

    v8f ut = tile_load16(upd + (size_t)b * DD * DD, DD, m0, n0);
    v8f qt = tile_load16(Q   + (size_t)b * DD * DD, DD, m0, n0);
    v8f nq;
#pragma unroll
    for (int r = 0; r < 8; ++r) {
        const float hc = tanhf(ah[r]);
        nq[r] = (1.0f - ut[r]) * qt[r] + ut[r] * hc;
    }
    tile_store16(newQ + (size_t)b * DD * DD, DD, m0, n0, nq);
}

// ---------------------------------------------------------------------------
// 8) generic batched WMMA GEMM, N fixed at 128 (8 waves), with epilogue modes:
//    mode 0: D = A@B ; mode 1: D = relu(A@B) ; mode 2: D = 0.5*(Add + relu(A@B))
//    grid = B * (M/16), block = 256
// ---------------------------------------------------------------------------
__global__ __launch_bounds__(256)
void gemm_n128_kernel(const float* __restrict__ A, int lda, size_t aStr,
                      const float* __restrict__ Bm, size_t bStr,
                      float* __restrict__ Dst, size_t dStr,
                      int M, int Kdim, int mode,
                      const float* __restrict__ Add, size_t addStr) {
    const int tilesM = M >> 4;
    const int b  = blockIdx.x / tilesM;
    const int m0 = (blockIdx.x % tilesM) * 16;
    const int n0 = (threadIdx.x >> 5) * 16;
    const float* __restrict__ Ab = A  + (size_t)b * aStr;
    const float* __restrict__ Bb = Bm + (size_t)b * bStr;
    v8f acc = {};
    wmma_kloop_f32(Ab, lda, Bb, DD, m0, n0, Kdim, acc);
    if (mode == 1) {
#pragma unroll
        for (int r = 0; r < 8; ++r) acc[r] = fmaxf(acc[r], 0.0f);
    } else if (mode == 2) {
        v8f h = tile_load16(Add + (size_t)b * addStr, DD, m0, n0);
#pragma unroll
        for (int r = 0; r < 8; ++r) acc[r] = 0.5f * (h[r] + fmaxf(acc[r], 0.0f));
    }
    tile_store16(Dst + (size_t)b * dStr, DD, m0, n0, acc);
}

// ---------------------------------------------------------------------------
// Host-side launch
// ---------------------------------------------------------------------------
extern "C" void kernel_launch(void* const* d_in, const int* in_sizes, int n_in,
                              void* d_out, int out_size, void* d_ws, size_t ws_size,
                              hipStream_t stream) {
    (void)in_sizes; (void)n_in; (void)out_size; (void)ws_size;
    const float* Ahat      = (const float*)d_in[0];
    const float* node_embs = (const float*)d_in[1];
    // d_in[2] = mask (unused by reference output)
    const float* ht        = (const float*)d_in[3];
    const float* prev_Q    = (const float*)d_in[4];
    const float* W_map     = (const float*)d_in[5];
    const float* b_map     = (const float*)d_in[6];
    const float* Wu        = (const float*)d_in[7];
    const float* Uu        = (const float*)d_in[8];
    const float* bu        = (const float*)d_in[9];
    const float* Wr        = (const float*)d_in[10];
    const float* Ur        = (const float*)d_in[11];
    const float* br        = (const float*)d_in[12];
    const float* Wh        = (const float*)d_in[13];
    const float* Uh        = (const float*)d_in[14];
    const float* bh        = (const float*)d_in[15];
    const float* static_W  = (const float*)d_in[16];

    float* out = (float*)d_out;
    // d_out layout: out[8*64*128] | policy[8] | scorer[8*128] | entropy[8]
    float* out_main    = out;
    float* out_policy  = out + (size_t)BB * TK * DD;
    float* out_scorer  = out_policy + BB;
    float* out_entropy = out_scorer + (size_t)BB * DD;

    float* w = (float*)d_ws;
    size_t off = 0;
    float* ws_scorer = w + off; off += (size_t)BB * DD;       // 1024
    float* ws_invn   = w + off; off += BB;                    // 8
    float* ws_scores = w + off; off += (size_t)BB * NN;       // 32768
    float* ws_stats  = w + off; off += 2 * BB;                // 16 (M, lse)
    int*   ws_tk     = (int*)(w + off); off += (size_t)BB * TK; // 512
    float* ws_z      = w + off; off += (size_t)BB * DD * DD;  // 131072
    float* ws_upd    = w + off; off += (size_t)BB * DD * DD;
    float* ws_rq     = w + off; off += (size_t)BB * DD * DD;
    float* ws_newq   = w + off; off += (size_t)BB * DD * DD;
    float* ws_ne     = w + off; off += (size_t)BB * TK * DD;  // 65536
    float* ws_A      = w + off; off += (size_t)BB * TK * TK;  // 32768
    float* ws_P      = w + off; off += (size_t)BB * TK * DD;
    float* ws_h1     = w + off; off += (size_t)BB * TK * DD;
    float* ws_S      = w + off; off += (size_t)BB * TK * DD;

    // 1) scorer + inv-norm
    scorer_kernel<<<BB, 128, 0, stream>>>(ht, W_map, b_map, ws_scorer, out_scorer, ws_invn);
    // 2) scores (normalized)
    scores_kernel<<<(BB * NN) / 8, 256, 0, stream>>>(node_embs, ws_scorer, ws_invn, ws_scores);
    // 3) softmax stats + entropy
    stats_kernel<<<BB, 256, 0, stream>>>(ws_scores, ws_stats, out_entropy);
    // 4) top-64 + policy score
    topk_kernel<<<BB, 256, 0, stream>>>(ws_scores, ws_stats, ws_tk, out_policy);
    // 5) gathers
    gather_zne_kernel<<<BB, 256, 0, stream>>>(node_embs, ws_scores, ws_tk, ws_z, ws_ne);
    gather_A_kernel<<<BB, 256, 0, stream>>>(Ahat, ws_tk, ws_A);
    // 6) GRU gates (update, reset*Q)
    gru_gate_kernel<<<BB * 8, 256, 0, stream>>>(Wu, Uu, bu, Wr, Ur, br, ws_z, prev_Q,
                                                ws_upd, ws_rq);
    // 7) GRU output (new_Q)
    gru_out_kernel<<<BB * 8, 256, 0, stream>>>(Wh, Uh, bh, ws_z, ws_rq, ws_upd, prev_Q,
                                               ws_newq);
    // 8) GCN: P = ne @ new_Q   (M=64, K=128)
    gemm_n128_kernel<<<BB * 4, 256, 0, stream>>>(ws_ne, DD, (size_t)TK * DD,
                                                 ws_newq, (size_t)DD * DD,
                                                 ws_P, (size_t)TK * DD,
                                                 TK, DD, 0, nullptr, 0);
    //    h1 = relu(A @ P)      (M=64, K=64)
    gemm_n128_kernel<<<BB * 4, 256, 0, stream>>>(ws_A, TK, (size_t)TK * TK,
                                                 ws_P, (size_t)TK * DD,
                                                 ws_h1, (size_t)TK * DD,
                                                 TK, TK, 1, nullptr, 0);
    //    S = h1 @ static_W     (M=64, K=128, B unbatched)
    gemm_n128_kernel<<<BB * 4, 256, 0, stream>>>(ws_h1, DD, (size_t)TK * DD,
                                                 static_W, 0,
                                                 ws_S, (size_t)TK * DD,
                                                 TK, DD, 0, nullptr, 0);
    //    out = 0.5*(h1 + relu(A @ S))  (M=64, K=64) -> d_out directly
    gemm_n128_kernel<<<BB * 4, 256, 0, stream>>>(ws_A, TK, (size_t)TK * TK,
                                                 ws_S, (size_t)TK * DD,
                                                 out_main, (size_t)TK * DD,
                                                 TK, TK, 2, ws_h1, (size_t)TK * DD);
}